// Subtractor50Bit_41085657153679
// MI455X (gfx1250) — compile-verified
//
#include <hip/hip_runtime.h>
#include <stdint.h>

#define BITS 50
#define NT   256                       // threads per block (8 waves)
#define RPW  32                        // rows per wave (1 row per lane)
#define ROWS (NT / 32 * RPW)           // 256 rows per block
#define WFLOATS (RPW * BITS)           // 1600 floats per operand per wave window
#define WBYTES  (WFLOATS * 4)          // 6400 bytes
#define CHUNK   (ROWS * BITS)          // 12800 floats per operand per block
#define WR128 12                       // 12 * 32 lanes * 16B = 6144 B
#define WR128_BYTES (WR128 * 32 * 16)  // 6144; remainder 256 B = 32 lanes x b64

typedef int v4i __attribute__((vector_size(16)));
typedef int v2i __attribute__((vector_size(8)));

typedef __attribute__((address_space(1))) v4i g_v4i;
typedef __attribute__((address_space(3))) v4i l_v4i;
typedef __attribute__((address_space(1))) v2i g_v2i;
typedef __attribute__((address_space(3))) v2i l_v2i;

#if defined(__has_builtin)
#if __has_builtin(__builtin_amdgcn_global_load_async_to_lds_b128) && \
    __has_builtin(__builtin_amdgcn_global_load_async_to_lds_b64)  && \
    __has_builtin(__builtin_amdgcn_global_store_async_from_lds_b128) && \
    __has_builtin(__builtin_amdgcn_global_store_async_from_lds_b64)
#define HAVE_ASYNC_LDS 1
#endif
#endif

__device__ __forceinline__ g_v4i* g4(const void* p) { return (g_v4i*)(uintptr_t)p; }
__device__ __forceinline__ g_v2i* g2(const void* p) { return (g_v2i*)(uintptr_t)p; }
__device__ __forceinline__ l_v4i* l4(void* p) { return (l_v4i*)p; }
__device__ __forceinline__ l_v2i* l2(void* p) { return (l_v2i*)p; }

__device__ __forceinline__ void wait_async_zero() {
#if defined(__has_builtin)
#if __has_builtin(__builtin_amdgcn_s_wait_asynccnt)
  __builtin_amdgcn_s_wait_asynccnt(0);
#else
  asm volatile("s_wait_asynccnt 0" ::: "memory");
#endif
#else
  asm volatile("s_wait_asynccnt 0" ::: "memory");
#endif
}

__device__ __forceinline__ void wait_ds_zero() {
#if defined(__has_builtin)
#if __has_builtin(__builtin_amdgcn_s_wait_dscnt)
  __builtin_amdgcn_s_wait_dscnt(0);
#else
  asm volatile("s_wait_dscnt 0" ::: "memory");
#endif
#else
  asm volatile("s_wait_dscnt 0" ::: "memory");
#endif
}

extern "C" __global__ void __launch_bounds__(NT)
Subtractor50Bit_kernel(const float* __restrict__ A, const float* __restrict__ B,
                       float* __restrict__ out, int N) {
  extern __shared__ float smem[];
  float* sA = smem;             // CHUNK floats; reused as result staging
  float* sB = smem + CHUNK;     // CHUNK floats

  const int t = threadIdx.x;
  const int w = t >> 5;                                   // wave id in block
  const int l = t & 31;                                   // lane id
  const long long base  = (long long)blockIdx.x * ROWS;   // first row of block
  const long long ebase = base * BITS;                    // first flat element
  float* __restrict__ bor = out + (long long)N * BITS;    // borrow section

  const bool full = (base + ROWS) <= (long long)N;

  if (full) {
    // -------- wave-private windows: no inter-wave barriers needed --------
    const char* gA = (const char*)(A + ebase) + w * WBYTES;
    const char* gB = (const char*)(B + ebase) + w * WBYTES;
    char* lA = (char*)sA + w * WBYTES;
    char* lB = (char*)sB + w * WBYTES;

#ifdef HAVE_ASYNC_LDS
#pragma unroll
    for (int k = 0; k < WR128; ++k) {
      const int off = (k * 32 + l) * 16;
      __builtin_amdgcn_global_load_async_to_lds_b128(g4(gA + off), l4(lA + off), 0, 0);
      __builtin_amdgcn_global_load_async_to_lds_b128(g4(gB + off), l4(lB + off), 0, 0);
    }
    {
      const int off = WR128_BYTES + l * 8;
      __builtin_amdgcn_global_load_async_to_lds_b64(g2(gA + off), l2(lA + off), 0, 0);
      __builtin_amdgcn_global_load_async_to_lds_b64(g2(gB + off), l2(lB + off), 0, 0);
    }
    wait_async_zero();   // wave's own LDS windows are now valid
#else
    {
      const float4* gA4 = (const float4*)gA;
      const float4* gB4 = (const float4*)gB;
      float4* lA4 = (float4*)lA;
      float4* lB4 = (float4*)lB;
#pragma unroll
      for (int k = 0; k < WR128; ++k) {
        const int i = k * 32 + l;
        lA4[i] = gA4[i];
        lB4[i] = gB4[i];
      }
      ((float2*)(lA + WR128_BYTES))[l] = ((const float2*)(gA + WR128_BYTES))[l];
      ((float2*)(lB + WR128_BYTES))[l] = ((const float2*)(gB + WR128_BYTES))[l];
    }
#endif

    // ------- pack bits -> u64, two's-complement subtract, unpack -------
    // A lane's row is 50 consecutive floats at LDS byte offset l*200 (8-byte
    // aligned), so access it as 25 float2's -> ds_load_b64 / ds_store_b64.
    float2* rowA2 = (float2*)(lA + l * (BITS * 4));
    float2* rowB2 = (float2*)(lB + l * (BITS * 4));
    unsigned alo = 0, ahi = 0, blo = 0, bhi = 0;
#pragma unroll
    for (int j = 0; j < 16; ++j) {       // bits 0..31
      const float2 a = rowA2[j];
      const float2 b = rowB2[j];
      // values are exactly 0.0f / 1.0f: bit pattern 0 or 0x3F800000 (< 2^30)
      alo |= (__float_as_uint(a.x) >> 29) << (2 * j);
      alo |= (__float_as_uint(a.y) >> 29) << (2 * j + 1);
      blo |= (__float_as_uint(b.x) >> 29) << (2 * j);
      blo |= (__float_as_uint(b.y) >> 29) << (2 * j + 1);
    }
#pragma unroll
    for (int j = 16; j < 25; ++j) {      // bits 32..49
      const float2 a = rowA2[j];
      const float2 b = rowB2[j];
      ahi |= (__float_as_uint(a.x) >> 29) << (2 * j - 32);
      ahi |= (__float_as_uint(a.y) >> 29) << (2 * j - 31);
      bhi |= (__float_as_uint(b.x) >> 29) << (2 * j - 32);
      bhi |= (__float_as_uint(b.y) >> 29) << (2 * j - 31);
    }
    const uint64_t av = (uint64_t)alo | ((uint64_t)ahi << 32);
    const uint64_t bv = (uint64_t)blo | ((uint64_t)bhi << 32);
    const uint64_t M   = (((uint64_t)1) << BITS) - 1;
    const uint64_t sum = av + ((~bv) & M) + 1;        // A + NOT(B) + 1
    const unsigned slo = (unsigned)sum;
    const unsigned shi = (unsigned)(sum >> 32);
#pragma unroll
    for (int j = 0; j < 16; ++j) {       // own row only: no cross-lane hazard
      float2 r;
      r.x = (float)((slo >> (2 * j)) & 1u);
      r.y = (float)((slo >> (2 * j + 1)) & 1u);
      rowA2[j] = r;
    }
#pragma unroll
    for (int j = 16; j < 25; ++j) {
      float2 r;
      r.x = (float)((shi >> (2 * j - 32)) & 1u);
      r.y = (float)((shi >> (2 * j - 31)) & 1u);
      rowA2[j] = r;
    }
    bor[base + w * RPW + l] = 1.0f - (float)((shi >> (BITS - 32)) & 1u);

    // ---------------- stream results back out (coalesced) ----------------
    char* gO = (char*)(out + ebase) + w * WBYTES;
#ifdef HAVE_ASYNC_LDS
    wait_ds_zero();  // order this wave's ds_stores before the DMA reads LDS
#pragma unroll
    for (int k = 0; k < WR128; ++k) {
      const int off = (k * 32 + l) * 16;
      __builtin_amdgcn_global_store_async_from_lds_b128(g4(gO + off), l4(lA + off), 0, 0);
    }
    {
      const int off = WR128_BYTES + l * 8;
      __builtin_amdgcn_global_store_async_from_lds_b64(g2(gO + off), l2(lA + off), 0, 0);
    }
    // S_ENDPGM performs an implicit wait-idle: ASYNCcnt drains before exit.
#else
    {
      float4* gO4 = (float4*)gO;
      const float4* lS4 = (const float4*)lA;
#pragma unroll
      for (int k = 0; k < WR128; ++k) {
        const int i = k * 32 + l;
        gO4[i] = lS4[i];
      }
      ((float2*)(gO + WR128_BYTES))[l] = ((const float2*)(lA + WR128_BYTES))[l];
    }
#endif
  } else {
    // ------------------------- tail block (N % 256) -------------------------
    const long long elems = ((long long)N - base) * BITS;
    for (int i = t; i < elems; i += NT) {
      sA[i] = A[ebase + i];
      sB[i] = B[ebase + i];
    }
    __syncthreads();
    if (base + t < (long long)N) {
      const int rb = t * BITS;
      uint64_t av = 0, bv = 0;
#pragma unroll
      for (int k = 0; k < BITS; ++k) {
        av |= (uint64_t)(__float_as_uint(sA[rb + k]) >> 29) << k;
        bv |= (uint64_t)(__float_as_uint(sB[rb + k]) >> 29) << k;
      }
      const uint64_t M   = (((uint64_t)1) << BITS) - 1;
      const uint64_t sum = av + ((~bv) & M) + 1;
#pragma unroll
      for (int k = 0; k < BITS; ++k)
        sA[rb + k] = (float)((sum >> k) & 1ull);
      bor[base + t] = 1.0f - (float)((sum >> BITS) & 1ull);
    }
    __syncthreads();
    for (int i = t; i < elems; i += NT) out[ebase + i] = sA[i];
  }
}

extern "C" void kernel_launch(void* const* d_in, const int* in_sizes, int n_in,
                              void* d_out, int out_size, void* d_ws, size_t ws_size,
                              hipStream_t stream) {
  (void)n_in; (void)out_size; (void)d_ws; (void)ws_size;
  const float* A = (const float*)d_in[0];
  const float* B = (const float*)d_in[1];
  float* out = (float*)d_out;
  const int N = in_sizes[0] / BITS;               // rows
  const int blocks = (N + ROWS - 1) / ROWS;
  const size_t shmem = (size_t)2 * CHUNK * sizeof(float);  // 100 KB (<=320 KB/WGP)
  Subtractor50Bit_kernel<<<blocks, NT, shmem, stream>>>(A, B, out, N);
}